// AcRKNContextLayer_71330816852079
// MI455X (gfx1250) — compile-verified
//
#include <hip/hip_runtime.h>
#include <hip/hip_bf16.h>
#include <math.h>

#define B_   512
#define T_   256
#define LOD_ 64
#define LAD_ 32
#define LTD_ 64
#define LSD_ 128
#define K_   15
#define CH_  15
#define NH_  15

typedef __attribute__((ext_vector_type(2))) float v2f;
typedef __attribute__((ext_vector_type(8))) float v8f;

// fp32 WMMA 16x16x4 (D = A(16x4) * B(4x16) + C), wave32.
// 8-arg form: (neg_a, A, neg_b, B, c_mod, C, reuse_a, reuse_b)
__device__ __forceinline__ v8f wmma_f32_16x16x4(v2f a, v2f b, v8f c) {
    return __builtin_amdgcn_wmma_f32_16x16x4_f32(false, a, false, b, (short)0, c,
                                                 false, false);
}

// ---------------------------------------------------------------------------
// Kernel A: per-batch softmax mixture coefficients (padded to 16 cols) and
// process-noise MLP  pn = elu(relu(task@nw1+nb1)@nw2+nb2)+1.
// ---------------------------------------------------------------------------
__global__ void prep_coeff_pnoise(const float* __restrict__ task,
                                  const float* __restrict__ w_coeff,
                                  const float* __restrict__ b_coeff,
                                  const float* __restrict__ nw1,
                                  const float* __restrict__ nb1,
                                  const float* __restrict__ nw2,
                                  const float* __restrict__ nb2,
                                  float* __restrict__ coeff16,
                                  float* __restrict__ pn) {
    int b = blockIdx.x * blockDim.x + threadIdx.x;
    if (b >= B_) return;
    float tk[LTD_];
    for (int j = 0; j < LTD_; ++j) tk[j] = task[b * LTD_ + j];

    float lg[K_];
    float mx = -1e30f;
    for (int k = 0; k < K_; ++k) {
        float s = b_coeff[k];
        for (int j = 0; j < LTD_; ++j) s += tk[j] * w_coeff[j * K_ + k];
        lg[k] = s;
        mx = fmaxf(mx, s);
    }
    float den = 0.f;
    for (int k = 0; k < K_; ++k) { lg[k] = __expf(lg[k] - mx); den += lg[k]; }
    float inv = 1.f / den;
    for (int k = 0; k < K_; ++k) coeff16[b * 16 + k] = lg[k] * inv;
    coeff16[b * 16 + 15] = 0.f;   // K padding for WMMA

    float h[NH_];
    for (int c = 0; c < NH_; ++c) {
        float s = nb1[c];
        for (int j = 0; j < LTD_; ++j) s += tk[j] * nw1[j * NH_ + c];
        h[c] = fmaxf(s, 0.f);
    }
    for (int o = 0; o < LSD_; ++o) {
        float s = nb2[o];
        for (int c = 0; c < NH_; ++c) s += h[c] * nw2[c * LSD_ + o];
        pn[b * LSD_ + o] = (s > 0.f) ? (s + 1.f) : __expf(s);   // elu + 1
    }
}

// ---------------------------------------------------------------------------
// Kernel B: pack the 4 basis tensors into band layout (K=16 rows x 2048 cols).
// Column c = m*512 + i*8 + d  holds tm_m[k][i][i+d-3] (0 outside band / d==7).
// ---------------------------------------------------------------------------
__global__ void pack_tm(const float* __restrict__ tm11,
                        const float* __restrict__ tm12,
                        const float* __restrict__ tm21,
                        const float* __restrict__ tm22,
                        float* __restrict__ tmpack) {
    int tid = blockIdx.x * blockDim.x + threadIdx.x;
    if (tid >= 16 * 2048) return;
    int k = tid >> 11;
    int c = tid & 2047;
    int m = c >> 9;
    int rem = c & 511;
    int i = rem >> 3;
    int d = rem & 7;
    int j = i + d - 3;
    float v = 0.f;
    if (k < K_ && d < 7 && j >= 0 && j < LOD_) {
        const float* tm = (m == 0) ? tm11 : (m == 1) ? tm12 : (m == 2) ? tm21 : tm22;
        v = tm[(k * LOD_ + i) * LOD_ + j];
    }
    tmpack[k * 2048 + c] = v;
}

// ---------------------------------------------------------------------------
// Kernel C: mix GEMM with fp32 WMMA:  tband(512 x 2048) = coeff16 @ tmpack.
// One wave per 16x16 tile; 4 k-steps of V_WMMA_F32_16X16X4_F32.
// Grid sized exactly (4096 waves) -> EXEC all-ones for WMMA.
// ---------------------------------------------------------------------------
__global__ void __launch_bounds__(128) mix_wmma(const float* __restrict__ coeff16,
                                                const float* __restrict__ tmpack,
                                                float* __restrict__ tband) {
    int wave = blockIdx.x * 4 + (threadIdx.x >> 5);
    int L = threadIdx.x & 31;
    int tn = wave & 127;          // 128 N tiles
    int tm = wave >> 7;           // 32 M tiles
    int m0 = tm << 4, n0 = tn << 4;
    int col = L & 15;
    int hi2 = (L >> 4) << 1;      // K sub-offset per half-wave (A/B layout)

    v8f c = {};
    #pragma unroll
    for (int k0 = 0; k0 < 16; k0 += 4) {
        v2f a, b;
        a.x = coeff16[(m0 + col) * 16 + k0 + hi2];
        a.y = coeff16[(m0 + col) * 16 + k0 + hi2 + 1];
        b.x = tmpack[(k0 + hi2) * 2048 + n0 + col];
        b.y = tmpack[(k0 + hi2 + 1) * 2048 + n0 + col];
        c = wmma_f32_16x16x4(a, b, c);
    }
    int rbase = m0 + ((L >> 4) << 3);
    #pragma unroll
    for (int r = 0; r < 8; ++r)
        tband[(size_t)(rbase + r) * 2048 + n0 + col] = c[r];
}

// ---------------------------------------------------------------------------
// Kernel D: expand per-batch mixed bands into the 14 time-invariant banded
// operators used by the scan (T11,T12,T21,T22 + 10 elementwise products).
// ---------------------------------------------------------------------------
__global__ void band_products(const float* __restrict__ tband,
                              float* __restrict__ wband) {
    int tid = blockIdx.x * blockDim.x + threadIdx.x;
    if (tid >= B_ * 512) return;
    int b = tid >> 9;
    int e = tid & 511;  // i*8 + d
    const float* tb = tband + (size_t)b * 2048;
    float t11 = tb[e], t12 = tb[512 + e], t21 = tb[1024 + e], t22 = tb[1536 + e];
    float* w = wband + (size_t)b * 7168;
    w[0 * 512 + e] = t11;
    w[1 * 512 + e] = t12;
    w[2 * 512 + e] = t21;
    w[3 * 512 + e] = t22;
    w[4 * 512 + e] = t11 * t11;
    w[5 * 512 + e] = t11 * t12;
    w[6 * 512 + e] = t12 * t12;
    w[7 * 512 + e] = t21 * t21;
    w[8 * 512 + e] = t21 * t22;
    w[9 * 512 + e] = t22 * t22;
    w[10 * 512 + e] = t21 * t11;
    w[11 * 512 + e] = t22 * t11;
    w[12 * 512 + e] = t21 * t12;
    w[13 * 512 + e] = t22 * t12;
}

// ---------------------------------------------------------------------------
// Kernel E: control MLP with fp32 WMMA.
//   ctrl = relu([action|task] @ cw1 + cb1) @ cw2 + cb2     (96 -> 15 -> 128)
// One wave per 16 rows; layer-1 K=96 (24 wmma), LDS bounce to re-stripe the
// hidden tile into A-fragments, layer-2 N=128 (8 tiles x 4 wmma).
// Grid sized exactly (8192 waves) -> EXEC all-ones.
// ---------------------------------------------------------------------------
__global__ void __launch_bounds__(128) control_mlp_wmma(
    const float* __restrict__ action, const float* __restrict__ task,
    const float* __restrict__ cw1, const float* __restrict__ cb1,
    const float* __restrict__ cw2, const float* __restrict__ cb2,
    float* __restrict__ ctrl) {
    __shared__ float hbuf[4][16][17];
    int wv = threadIdx.x >> 5;
    int L = threadIdx.x & 31;
    int wave = blockIdx.x * 4 + wv;
    int r0 = wave << 4;                 // flat row r = b*T + t
    int col = L & 15;
    int hi2 = (L >> 4) << 1;
    int rofs = (L >> 4) << 3;
    int rowA = r0 + col;                // A-fragment: lane -> M
    int bA = rowA >> 8;                 // T_ == 256

    v8f c1 = {};
    for (int k0 = 0; k0 < 96; k0 += 4) {
        int kx = k0 + hi2;
        v2f a, b;
        a.x = (kx < LAD_) ? action[(size_t)rowA * LAD_ + kx]
                          : task[(size_t)bA * LTD_ + kx - LAD_];
        a.y = (kx + 1 < LAD_) ? action[(size_t)rowA * LAD_ + kx + 1]
                              : task[(size_t)bA * LTD_ + kx + 1 - LAD_];
        b.x = (col < CH_) ? cw1[kx * CH_ + col] : 0.f;
        b.y = (col < CH_) ? cw1[(kx + 1) * CH_ + col] : 0.f;
        c1 = wmma_f32_16x16x4(a, b, c1);
    }
    float bias1 = (col < CH_) ? cb1[col] : 0.f;
    #pragma unroll
    for (int r = 0; r < 8; ++r) {
        float h = fmaxf(c1[r] + bias1, 0.f);
        if (col >= CH_) h = 0.f;        // keep K-padding exactly zero
        hbuf[wv][rofs + r][col] = h;
    }
    __syncthreads();

    #pragma unroll
    for (int nt = 0; nt < 8; ++nt) {
        int n0 = nt << 4;
        v8f c2 = {};
        #pragma unroll
        for (int k0 = 0; k0 < 16; k0 += 4) {
            int kx = k0 + hi2;
            v2f a, b;
            a.x = hbuf[wv][col][kx];
            a.y = hbuf[wv][col][kx + 1];
            b.x = (kx < CH_) ? cw2[kx * LSD_ + n0 + col] : 0.f;
            b.y = (kx + 1 < CH_) ? cw2[(kx + 1) * LSD_ + n0 + col] : 0.f;
            c2 = wmma_f32_16x16x4(a, b, c2);
        }
        float bias2 = cb2[n0 + col];
        #pragma unroll
        for (int r = 0; r < 8; ++r) {
            int row = r0 + rofs + r;
            ctrl[(size_t)row * LSD_ + n0 + col] = c2[r] + bias2;
        }
    }
}

// ---------------------------------------------------------------------------
// Kernel F: the sequential scan. One workgroup (64 threads, 2 waves) per
// batch; the 14 banded operators (28 KB) + halo-padded share vectors live in
// LDS; carry state lives in registers; next-step inputs are prefetched.
// ---------------------------------------------------------------------------
__global__ void __launch_bounds__(64) acrkn_scan_kernel(
    const float* __restrict__ latent_obs,
    const float* __restrict__ obs_vars,
    const unsigned char* __restrict__ obs_valid,
    const float* __restrict__ initial_mean,
    const float* __restrict__ initial_cov,
    const float* __restrict__ ctrl,
    const float* __restrict__ pn,
    const float* __restrict__ wband,
    float* __restrict__ out) {
    __shared__ float sband[14 * 512];
    __shared__ float s_mu_u[70], s_mu_l[70], s_pcu[70], s_pcl[70], s_pcs[70];

    int b = blockIdx.x;
    int i = threadIdx.x;

    const float* wb = wband + (size_t)b * 7168;
    for (int idx = i; idx < 7168; idx += 64) sband[idx] = wb[idx];
    for (int idx = i; idx < 70; idx += 64) {  // zero halos (interior rewritten)
        s_mu_u[idx] = 0.f; s_mu_l[idx] = 0.f;
        s_pcu[idx] = 0.f; s_pcl[idx] = 0.f; s_pcs[idx] = 0.f;
    }

    float pm_u = initial_mean[(size_t)b * LSD_ + i];
    float pm_l = initial_mean[(size_t)b * LSD_ + LOD_ + i];
    float cu = initial_cov[(size_t)b * 3 * LOD_ + i];
    float cl = initial_cov[(size_t)b * 3 * LOD_ + LOD_ + i];
    float cs = initial_cov[(size_t)b * 3 * LOD_ + 2 * LOD_ + i];
    float pnu = pn[(size_t)b * LSD_ + i];
    float pnl = pn[(size_t)b * LSD_ + LOD_ + i];
    __syncthreads();

    const size_t OFF_PC = (size_t)B_ * T_ * LSD_;
    const size_t OFF_QM = OFF_PC + (size_t)B_ * T_ * 3 * LOD_;
    const size_t OFF_QC = OFF_QM + (size_t)B_ * T_ * LSD_;

    for (int t = 0; t < T_; ++t) {
        int bt = b * T_ + t;
        float obs = latent_obs[(size_t)bt * LOD_ + i];
        float var = obs_vars[(size_t)bt * LOD_ + i];
        float clo = ctrl[(size_t)bt * LSD_ + i];
        float chi = ctrl[(size_t)bt * LSD_ + LOD_ + i];
        bool valid = obs_valid[bt] != 0;
        if (t + 1 < T_) {   // hide next-step latency (global_prefetch_b8)
            __builtin_prefetch(&latent_obs[(size_t)(bt + 1) * LOD_ + i], 0, 3);
            __builtin_prefetch(&obs_vars[(size_t)(bt + 1) * LOD_ + i], 0, 3);
            __builtin_prefetch(&ctrl[(size_t)(bt + 1) * LSD_ + i], 0, 3);
        }

        // --- Kalman update (elementwise, factorized covariance) ---
        float denom = cu + var;
        float q_u = cu / denom;
        float q_l = cs / denom;
        float resid = obs - pm_u;
        float post_u = pm_u + q_u * resid;
        float post_l = pm_l + q_l * resid;
        float p_cu = (1.f - q_u) * cu;
        float p_cl = cl - q_l * cs;
        float p_cs = (1.f - q_u) * cs;
        if (!valid) {
            post_u = pm_u; post_l = pm_l;
            p_cu = cu; p_cl = cl; p_cs = cs;
        }

        out[(size_t)bt * LSD_ + i] = post_u;
        out[(size_t)bt * LSD_ + LOD_ + i] = post_l;
        out[OFF_PC + (size_t)bt * 192 + i] = p_cu;
        out[OFF_PC + (size_t)bt * 192 + LOD_ + i] = p_cl;
        out[OFF_PC + (size_t)bt * 192 + 2 * LOD_ + i] = p_cs;

        s_mu_u[3 + i] = post_u; s_mu_l[3 + i] = post_l;
        s_pcu[3 + i] = p_cu; s_pcl[3 + i] = p_cl; s_pcs[3 + i] = p_cs;
        __syncthreads();

        // --- banded predict: 14 operators x 7 diagonals, branch-free ---
        float nmu_u = clo, nmu_l = chi;
        float ncu = pnu, ncl = pnl, ncs = 0.f;
        #pragma unroll
        for (int d = 0; d < 7; ++d) {
            float muu = s_mu_u[i + d], mul = s_mu_l[i + d];
            float pcu = s_pcu[i + d], pcl = s_pcl[i + d], pcs = s_pcs[i + d];
            int e = i * 8 + d;
            nmu_u += sband[0 * 512 + e] * muu + sband[1 * 512 + e] * mul;
            nmu_l += sband[2 * 512 + e] * muu + sband[3 * 512 + e] * mul;
            ncu += sband[4 * 512 + e] * pcu + 2.f * sband[5 * 512 + e] * pcs +
                   sband[6 * 512 + e] * pcl;
            ncl += sband[7 * 512 + e] * pcu + 2.f * sband[8 * 512 + e] * pcs +
                   sband[9 * 512 + e] * pcl;
            ncs += sband[10 * 512 + e] * pcu +
                   (sband[11 * 512 + e] + sband[12 * 512 + e]) * pcs +
                   sband[13 * 512 + e] * pcl;
        }

        out[OFF_QM + (size_t)bt * LSD_ + i] = nmu_u;
        out[OFF_QM + (size_t)bt * LSD_ + LOD_ + i] = nmu_l;
        out[OFF_QC + (size_t)bt * 192 + i] = ncu;
        out[OFF_QC + (size_t)bt * 192 + LOD_ + i] = ncl;
        out[OFF_QC + (size_t)bt * 192 + 2 * LOD_ + i] = ncs;

        pm_u = nmu_u; pm_l = nmu_l;
        cu = ncu; cl = ncl; cs = ncs;
        __syncthreads();   // protect shared vectors before next step's writes
    }
}

// ---------------------------------------------------------------------------
extern "C" void kernel_launch(void* const* d_in, const int* in_sizes, int n_in,
                              void* d_out, int out_size, void* d_ws, size_t ws_size,
                              hipStream_t stream) {
    const float* latent_obs = (const float*)d_in[0];
    const float* obs_vars = (const float*)d_in[1];
    const float* action = (const float*)d_in[2];
    const float* latent_task = (const float*)d_in[3];
    const float* initial_mean = (const float*)d_in[4];
    const float* initial_cov = (const float*)d_in[5];
    const unsigned char* obs_valid = (const unsigned char*)d_in[6];  // jnp.bool_
    const float* tm11 = (const float*)d_in[7];
    const float* tm12 = (const float*)d_in[8];
    const float* tm21 = (const float*)d_in[9];
    const float* tm22 = (const float*)d_in[10];
    const float* w_coeff = (const float*)d_in[11];
    const float* b_coeff = (const float*)d_in[12];
    const float* cw1 = (const float*)d_in[13];
    const float* cb1 = (const float*)d_in[14];
    const float* cw2 = (const float*)d_in[15];
    const float* cb2 = (const float*)d_in[16];
    const float* nw1 = (const float*)d_in[17];
    const float* nb1 = (const float*)d_in[18];
    const float* nw2 = (const float*)d_in[19];
    const float* nb2 = (const float*)d_in[20];

    float* w = (float*)d_ws;
    float* coeff16 = w;                      //   512*16      =      8192
    float* pn      = w + 8192;               //   512*128     =     65536
    float* tmpack  = w + 73728;              //    16*2048    =     32768
    float* tband   = w + 106496;             //   512*2048    =   1048576
    float* wband   = w + 1155072;            //   512*14*512  =   3670016
    float* ctrl    = w + 4825088;            //   512*256*128 =  16777216
    // total: 21602304 floats (~86.4 MB) of workspace

    float* out = (float*)d_out;

    prep_coeff_pnoise<<<4, 128, 0, stream>>>(latent_task, w_coeff, b_coeff,
                                             nw1, nb1, nw2, nb2, coeff16, pn);
    pack_tm<<<128, 256, 0, stream>>>(tm11, tm12, tm21, tm22, tmpack);
    mix_wmma<<<1024, 128, 0, stream>>>(coeff16, tmpack, tband);
    band_products<<<1024, 256, 0, stream>>>(tband, wband);
    control_mlp_wmma<<<2048, 128, 0, stream>>>(action, latent_task, cw1, cb1,
                                               cw2, cb2, ctrl);
    acrkn_scan_kernel<<<512, 64, 0, stream>>>(latent_obs, obs_vars, obs_valid,
                                              initial_mean, initial_cov, ctrl,
                                              pn, wband, out);
}